// EpplRender_36009005810212
// MI455X (gfx1250) — compile-verified
//
#include <hip/hip_runtime.h>
#include <hip/hip_bf16.h>

// Problem constants (B=1 hardcoded)
#define SN   4
#define H    96
#define W    320
#define SR   7
#define HW   (H * W)        // 30720
#define NPIX (SN * HW)      // 122880

// exp(-0.5*q/KWS^2) = exp2(q * EXP2SCALE), KWS = 2.3
// |q| <= ~900  =>  |arg| <= ~123 < 126: raw v_exp_f32 is safe, no range fixup needed.
__device__ __constant__ float EXP2SCALE = -0.5f / (2.3f * 2.3f) * 1.4426950408889634f;

typedef float v2f __attribute__((ext_vector_type(2)));
typedef float v8f __attribute__((ext_vector_type(8)));

__global__ void eppl_zero_kernel(float* ws, int n) {
    int i = blockIdx.x * blockDim.x + threadIdx.x;
    if (i < n) ws[i] = 0.0f;
}

// One wave (32 lanes) per source pixel.
// quad[m][n] = u[n] + fy[m]*v[n] + w[m]  computed as 16x4 @ 4x16 f32 WMMA:
//   A[m]   = [1, fy[m], M11*fy[m]^2, 0]              (16x4)
//   B[:,n] = [M00*fx[n]^2, (M01+M10)*fx[n], 1, 0]    (4x16)
__global__ void eppl_splat_kernel(const float* __restrict__ irs,   // (SN,H,W,2,2)
                                  const float* __restrict__ p2d,   // (SN,2,H,W)
                                  const float* __restrict__ sel,   // (1,H,W)
                                  float* __restrict__ acc,         // (SN,H,W)
                                  float* __restrict__ cnt) {       // (SN,H,W)
    const int wavesPerBlock = blockDim.x >> 5;
    const int wave = blockIdx.x * wavesPerBlock + (threadIdx.x >> 5);
    const int lane = threadIdx.x & 31;

    const int sn = wave / HW;
    const int hw = wave - sn * HW;

    // Wave-uniform scalar loads (all lanes read the same cacheline)
    const float px = p2d[(sn * 2 + 0) * HW + hw];
    const float py = p2d[(sn * 2 + 1) * HW + hw];
    const float* Mp = irs + (size_t)(sn * HW + hw) * 4;
    const float M00 = Mp[0];
    const float M01 = Mp[1];
    const float M10 = Mp[2];
    const float M11 = Mp[3];
    const float selv = sel[hw];

    const float cxf = rintf(px);          // round-half-even, matches jnp.round
    const float cyf = rintf(py);
    const int cx = (int)cxf;
    const int cy = (int)cyf;

    const int  mn = lane & 15;            // this lane's row (dy idx) AND col (dx idx)
    const bool hi = lane >= 16;

    const float fy = (cyf + (float)(mn - SR)) - py;
    const float fx = (cxf + (float)(mn - SR)) - px;

    // A operand: lanes 0-15 -> (A[m][0], A[m][1]); lanes 16-31 -> (A[m][2], A[m][3])
    v2f a;
    a.x = hi ? M11 * fy * fy : 1.0f;
    a.y = hi ? 0.0f          : fy;
    // B operand: lanes 0-15 -> (B[0][n], B[1][n]); lanes 16-31 -> (B[2][n], B[3][n])
    v2f b;
    b.x = hi ? 1.0f : M00 * fx * fx;
    b.y = hi ? 0.0f : (M01 + M10) * fx;

    v8f c = {};
    // D = A x B + 0 : 16x16 tile of quadratic-form values, one instruction
    v8f d = __builtin_amdgcn_wmma_f32_16x16x4_f32(
        /*neg_a=*/false, a, /*neg_b=*/false, b,
        /*c_mod=*/(short)0, c, /*reuse_a=*/false, /*reuse_b=*/false);

    // Evaluate all 8 weights up front: 8 independent v_mul + v_exp_f32 chains
    // (bare transcendental, argument range-proven; chains co-execute with VALU).
    float wgt[8];
#pragma unroll
    for (int r = 0; r < 8; ++r)
        wgt[r] = __builtin_amdgcn_exp2f(d[r] * EXP2SCALE);

    // This lane owns column n = mn of the D tile (rows r+8*hi for r=0..7)
    const int  txi   = cx + (mn - SR);
    const bool colok = (mn < 15) && (txi >= 0) && (txi < W) && (selv > 0.0f);

    float* accp = acc + (size_t)sn * HW;
    float* cntp = cnt + (size_t)sn * HW;

#pragma unroll
    for (int r = 0; r < 8; ++r) {
        const int  m   = r + (hi ? 8 : 0);       // dy index 0..15
        const int  tyi = cy + (m - SR);
        const bool ok  = colok && (m < 15) && (tyi >= 0) && (tyi < H);
        if (ok) {
            const int off = tyi * W + txi;
            atomicAdd(accp + off, wgt[r]);       // global_atomic_add_f32
            atomicAdd(cntp + off, 1.0f);
        }
    }
}

__global__ void eppl_finalize_kernel(const float* __restrict__ acc,
                                     const float* __restrict__ cnt,
                                     float* __restrict__ out, int n) {
    int i = blockIdx.x * blockDim.x + threadIdx.x;
    if (i < n) out[i] = acc[i] / fmaxf(cnt[i], 1.0f);
}

extern "C" void kernel_launch(void* const* d_in, const int* in_sizes, int n_in,
                              void* d_out, int out_size, void* d_ws, size_t ws_size,
                              hipStream_t stream) {
    const float* irs = (const float*)d_in[0];   // inv_r_sigma (1,SN,H,W,2,2)
    const float* p2d = (const float*)d_in[1];   // projected2d (1,SN,2,H,W)
    const float* sel = (const float*)d_in[2];   // selector    (1,1,H,W)
    float* out = (float*)d_out;

    float* acc = (float*)d_ws;                  // SN*H*W floats
    float* cnt = acc + NPIX;                    // SN*H*W floats (total < 1 MB)

    // 1) zero accumulators (ws is poisoned by the harness)
    eppl_zero_kernel<<<(2 * NPIX + 255) / 256, 256, 0, stream>>>(acc, 2 * NPIX);

    // 2) splat: one wave per source pixel, 8 waves per 256-thread block
    //    NPIX/8 = 15360 blocks exactly -> no partial waves, EXEC all-1s at WMMA
    eppl_splat_kernel<<<NPIX / 8, 256, 0, stream>>>(irs, p2d, sel, acc, cnt);

    // 3) out = acc / max(cnt, 1)
    eppl_finalize_kernel<<<(NPIX + 255) / 256, 256, 0, stream>>>(acc, cnt, out, NPIX);
}